// CaptionEmbedder_28845000360093
// MI455X (gfx1250) — compile-verified
//
#include <hip/hip_runtime.h>
#include <hip/hip_bf16.h>

// Problem constants (from the reference)
#define K_VOCAB  32000
#define K_NENT   64
#define K_NFACT  512
#define K_D      512     // floats per row  (2 KB)
#define K_L      128     // tokens per batch row

// One block = one token. 128 lanes * 16B = 512 floats = one row.
__global__ __launch_bounds__(128) void CaptionEmbedder_gather_kernel(
    const int*   __restrict__ cap_idx,   // [B*L]
    const float* __restrict__ ents,      // [B, N_ENT, D]
    const float* __restrict__ facts,     // [B, N_FACT, D]
    const float* __restrict__ words,     // [VOCAB, D]
    const int*   __restrict__ pad_tok,   // [1]
    const int*   __restrict__ masks,     // [B*L]
    float*       __restrict__ out)       // [B*L, D]
{
    __shared__ float smem[K_D];          // 2 KB staging, one 16B slot per lane

    const int token = blockIdx.x;        // 0 .. B*L-1   (uniform per block)
    const int b     = token / K_L;
    const int lane  = threadIdx.x;       // 0 .. 127

    // Uniform per-block control data -> scalar branch, scalarized base address.
    const int ci = cap_idx[token];
    const int m  = masks[token];

    const float* src;
    if (m == 2) {
        int fi = ci - (K_VOCAB + K_NENT);
        if (fi < 0 || fi >= K_NFACT) fi = K_NFACT - 1;
        src = facts + ((size_t)b * K_NFACT + (size_t)fi) * K_D;
    } else if (m == 1) {
        int ei = ci - K_VOCAB;
        if (ei < 0 || ei >= K_NENT) ei = K_NENT - 1;
        src = ents + ((size_t)b * K_NENT + (size_t)ei) * K_D;
    } else {
        int wi = (ci >= K_VOCAB) ? pad_tok[0] : ci;
        src = words + (size_t)wi * K_D;
    }

    const float* gsrc = src + 4 * lane;                        // 16B per lane
    float*       gdst = out + (size_t)token * K_D + 4 * lane;
    // LDS byte offset for this lane's private 16B slot (low 32 bits of the
    // generic shared-aperture address are the LDS offset on gfx1250).
    unsigned lds_off = (unsigned)(size_t)(&smem[4 * lane]);

    // CDNA5 async DMA: global -> LDS, 128 bits per lane (ASYNCcnt-tracked).
    asm volatile("global_load_async_to_lds_b128 %0, %1, off"
                 :: "v"(lds_off), "v"(gsrc)
                 : "memory");

    // Each lane only touches its own LDS slot written by its own wave's async
    // load, so a per-wave asynccnt wait is the only ordering needed.
#if __has_builtin(__builtin_amdgcn_s_wait_asynccnt)
    __builtin_amdgcn_s_wait_asynccnt(0);
#else
    asm volatile("s_wait_asynccnt 0" ::: "memory");
#endif

    // CDNA5 async DMA: LDS -> global, 128 bits per lane.
    asm volatile("global_store_async_from_lds_b128 %0, %1, off"
                 :: "v"(gdst), "v"(lds_off)
                 : "memory");

    // S_ENDPGM performs an implicit wait-idle; explicit wait kept for clarity.
#if __has_builtin(__builtin_amdgcn_s_wait_asynccnt)
    __builtin_amdgcn_s_wait_asynccnt(0);
#else
    asm volatile("s_wait_asynccnt 0" ::: "memory");
#endif
}

extern "C" void kernel_launch(void* const* d_in, const int* in_sizes, int n_in,
                              void* d_out, int out_size, void* d_ws, size_t ws_size,
                              hipStream_t stream) {
    (void)n_in; (void)out_size; (void)d_ws; (void)ws_size;

    const int*   cap_idx = (const int*)  d_in[0];   // [B*L] int32
    const float* ents    = (const float*)d_in[1];   // [B, N_ENT, D]
    const float* facts   = (const float*)d_in[2];   // [B, N_FACT, D]
    const float* words   = (const float*)d_in[3];   // [VOCAB, D]
    const int*   pad_tok = (const int*)  d_in[4];   // [1]
    const int*   masks   = (const int*)  d_in[5];   // [B*L]
    float*       out     = (float*)      d_out;     // [B*L, D]

    const int n_tokens = in_sizes[0];               // B*L = 16384

    CaptionEmbedder_gather_kernel<<<n_tokens, 128, 0, stream>>>(
        cap_idx, ents, facts, words, pad_tok, masks, out);
}